// ChunkMasker_29523605192737
// MI455X (gfx1250) — compile-verified
//
#include <hip/hip_runtime.h>
#include <stdint.h>

// ---------------------------------------------------------------------------
// ChunkMasker for MI455X (gfx1250, wave32).
// Memory-bound: ~258MB effective traffic / 23.3 TB/s ~= 11us floor.
// K0: parallel threefry2x32 draws.  K1: per-batch sequential span scan,
// one wave each, mask words register-resident (2/lane), uniform v_readlane
// fetches.  K2: the 295MB stream runs on the CDNA5 async data-mover:
// global_load_async_to_lds_b128 -> s_wait_asynccnt -> 
// global_store_async_from_lds_b128 (VGPR-free copy, masked chunks skip the
// load and store from a pre-zeroed LDS region).
// ---------------------------------------------------------------------------

#define MASK_B 32
#define MASK_T 1500
#define MASK_D 768
#define NUM_TO_MASK 375
#define GUARD 1500             // 4 * num_to_mask
#define NDRAW (GUARD * MASK_B) // 48000 draws per stream
#define NPAIR (NDRAW / 2)      // 24000 threefry blocks per stream
#define NWORDS 47              // ceil(1500/32) mask words per batch elem
#define WSTRIDE 48             // padded stride for packed mask words
#define X_ELEMS (MASK_B * MASK_T * MASK_D)   // 36,864,000 floats
#define N_F4 (X_ELEMS / 4)                   // 9,216,000 float4

#define CHUNK_F4 32                 // one b128 per lane -> 512B per chunk
#define CHUNKS_PER_ROW 6            // 768 floats / 128 floats-per-chunk
#define N_CH (N_F4 / CHUNK_F4)      // 288,000 chunks
#define CH 16                       // chunks per wave iteration (8KB staging)
#define N_SC (N_CH / CH)            // 18,000 superchunks (exact)
#define APPLY_WAVES_PER_BLOCK 4
#define LDS_STAGE_BYTES (APPLY_WAVES_PER_BLOCK * CH * 512)

typedef float f4 __attribute__((ext_vector_type(4)));

// ---- threefry2x32, 20 rounds (JAX's PRNG core) ----------------------------
__host__ __device__ inline void tf2x32(uint32_t k0, uint32_t k1,
                                       uint32_t x0, uint32_t x1,
                                       uint32_t& o0, uint32_t& o1) {
  uint32_t ks2 = 0x1BD11BDAu ^ k0 ^ k1;
  x0 += k0; x1 += k1;
#define TFR(r) { x0 += x1; x1 = (x1 << (r)) | (x1 >> (32 - (r))); x1 ^= x0; }
  TFR(13) TFR(15) TFR(26) TFR(6)   x0 += k1;  x1 += ks2 + 1u;
  TFR(17) TFR(29) TFR(16) TFR(24)  x0 += ks2; x1 += k0 + 2u;
  TFR(13) TFR(15) TFR(26) TFR(6)   x0 += k0;  x1 += k1 + 3u;
  TFR(17) TFR(29) TFR(16) TFR(24)  x0 += k1;  x1 += ks2 + 4u;
  TFR(13) TFR(15) TFR(26) TFR(6)   x0 += ks2; x1 += k0 + 5u;
#undef TFR
  o0 = x0; o1 = x1;
}

// ---- K0: generate all (len,start) pairs in parallel -----------------------
__device__ inline void emit_pair(uint32_t* rng, uint32_t i,
                                 uint32_t lenbits, uint32_t startbits) {
  uint32_t len = 2u + (lenbits & 3u);          // randint [2,6), pow2 span
  uint32_t span = (uint32_t)MASK_T - len;      // max(1, T-len) == T-len
  uint32_t start = startbits % span;           // randint [0, T-len)
  rng[i] = (len << 16) | start;
}

__global__ __launch_bounds__(256) void rng_kernel(uint32_t* __restrict__ rng,
                                                  uint32_t k1a, uint32_t k1b,
                                                  uint32_t k2a, uint32_t k2b) {
  uint32_t j = blockIdx.x * blockDim.x + threadIdx.x;
  if (j >= NPAIR) return;
  uint32_t l0, l1, s0, s1;
  tf2x32(k1a, k1b, j, j + NPAIR, l0, l1);
  tf2x32(k2a, k2b, j, j + NPAIR, s0, s1);
  emit_pair(rng, j,         l0, s0);
  emit_pair(rng, j + NPAIR, l1, s1);
}

// ---- K1: sequential span scan, one wave per batch element -----------------
__device__ inline uint32_t range_mask(uint32_t lo, uint32_t hi) {
  uint32_t m = (hi >= 32u) ? 0xFFFFFFFFu : ((1u << hi) - 1u);
  return m & (0xFFFFFFFFu << lo);
}

__global__ __launch_bounds__(32) void mask_kernel(const uint32_t* __restrict__ rng,
                                                  uint32_t* __restrict__ mw,
                                                  float* __restrict__ maskf) {
  const int b = blockIdx.x;
  const int lane = threadIdx.x;
  uint32_t wA = 0u, wB = 0u;   // word[lane], word[lane+32]
  int count = 0;

  for (int g = 0; g < GUARD; ++g) {
    uint32_t r = rng[g * MASK_B + b];            // uniform -> scalar load
    uint32_t start = r & 0xFFFFu;
    uint32_t len = r >> 16;
    uint32_t end = start + len;
    uint32_t w0 = start >> 5, b0 = start & 31u;
    uint32_t wL = (end - 1u) >> 5;
    bool two = (wL != w0);

    uint32_t hi0 = two ? 32u : (end - (w0 << 5));
    uint32_t m0 = range_mask(b0, hi0);
    uint32_t m1 = two ? range_mask(0u, end - (wL << 5)) : 0u;

    uint32_t v0 = __shfl((w0 >= 32u) ? wB : wA, (int)(w0 & 31u), 32);
    uint32_t v1 = __shfl((wL >= 32u) ? wB : wA, (int)(wL & 31u), 32);

    if (lane == (int)(w0 & 31u)) { if (w0 >= 32u) wB |= m0; else wA |= m0; }
    if (two && lane == (int)(wL & 31u)) { if (wL >= 32u) wB |= m1; else wA |= m1; }

    count += __popc(m0 & ~v0) + __popc(m1 & ~v1); // uniform across lanes
    if (count >= NUM_TO_MASK) break;              // uniform exit == done
  }

  __shared__ uint32_t sh[NWORDS];
  sh[lane] = wA;
  if (lane + 32 < NWORDS) sh[lane + 32] = wB;
  __syncthreads();

  mw[b * WSTRIDE + lane] = wA;
  if (lane + 32 < NWORDS) mw[b * WSTRIDE + lane + 32] = wB;

  for (int t = lane; t < MASK_T; t += 32)
    maskf[b * MASK_T + t] = ((sh[t >> 5] >> (t & 31)) & 1u) ? 1.0f : 0.0f;
}

// ---- K2: masked stream on the async data mover ----------------------------
// Each wave: 16 chunks (8KB contiguous).  Unmasked chunks: async load
// global->LDS; all chunks: async store LDS->global (masked ones from a
// pre-zeroed region).  Data never passes through VGPRs.
__global__ __launch_bounds__(APPLY_WAVES_PER_BLOCK * 32)
void apply_async_kernel(const f4* __restrict__ x4, f4* __restrict__ o4,
                        const uint32_t* __restrict__ mw) {
  __shared__ __align__(16) char lds[LDS_STAGE_BYTES + 512];
  const int lane = threadIdx.x & 31;
  const int wave = threadIdx.x >> 5;

  // block-shared zero chunk (written once, only ever read by async stores)
  char* zbuf = &lds[LDS_STAGE_BYTES];
  if (threadIdx.x < 32) {
    f4 z = {0.f, 0.f, 0.f, 0.f};
    *(f4*)(zbuf + threadIdx.x * 16) = z;
  }
  __syncthreads();

  char* wbuf = &lds[wave * (CH * 512)];
  const uint32_t zoff = (uint32_t)(uintptr_t)zbuf + (uint32_t)(lane * 16);

  const int nWaves = (gridDim.x * blockDim.x) >> 5;
  const int gwave = (int)((blockIdx.x * blockDim.x + threadIdx.x) >> 5);

  for (int sc = gwave; sc < N_SC; sc += nWaves) {
    // previous iteration's async stores must finish before buffer reuse
    asm volatile("s_wait_asynccnt 0x0" ::: "memory");

    const int c0 = sc * CH;
    uint32_t mbits = 0;
    for (int k = 0; k < CH; ++k) {
      int c = c0 + k;
      int row = c / CHUNKS_PER_ROW;            // mask is chunk-uniform
      int bb = row / MASK_T;
      int t = row - bb * MASK_T;
      uint32_t bit = (mw[bb * WSTRIDE + (t >> 5)] >> (t & 31)) & 1u;
      mbits |= bit << k;
      if (!bit) {
        uint32_t loff = (uint32_t)(uintptr_t)(wbuf + k * 512) + (uint32_t)(lane * 16);
        uint64_t ga = (uint64_t)(uintptr_t)(&x4[c * CHUNK_F4 + lane]);
        asm volatile("global_load_async_to_lds_b128 %0, %1, off"
                     :: "v"(loff), "v"(ga) : "memory");
      }
    }
    if (sc + nWaves < N_SC)                    // gfx1250 global_prefetch_b8
      __builtin_prefetch(&x4[(sc + nWaves) * CH * CHUNK_F4 + lane], 0, 0);

    // all loads landed in LDS (loads complete in order; cnt==0 => all done)
    asm volatile("s_wait_asynccnt 0x0" ::: "memory");

    for (int k = 0; k < CH; ++k) {
      int c = c0 + k;
      uint32_t loff = ((mbits >> k) & 1u)
          ? zoff
          : (uint32_t)(uintptr_t)(wbuf + k * 512) + (uint32_t)(lane * 16);
      uint64_t ga = (uint64_t)(uintptr_t)(&o4[c * CHUNK_F4 + lane]);
      asm volatile("global_store_async_from_lds_b128 %0, %1, off"
                   :: "v"(ga), "v"(loff) : "memory");
    }
  }
  // drain (S_ENDPGM also implies wait-idle, this is belt-and-braces)
  asm volatile("s_wait_asynccnt 0x0" ::: "memory");
}

// ---------------------------------------------------------------------------
extern "C" void kernel_launch(void* const* d_in, const int* in_sizes, int n_in,
                              void* d_out, int out_size, void* d_ws, size_t ws_size,
                              hipStream_t stream) {
  (void)in_sizes; (void)n_in; (void)out_size; (void)ws_size;

  // jax.random.split(key(1)): threefry over iota(4), half-split counters
  uint32_t a0, a1, b0, b1;
  tf2x32(0u, 1u, 0u, 2u, a0, a1);
  tf2x32(0u, 1u, 1u, 3u, b0, b1);
  const uint32_t k1a = a0, k1b = b0;
  const uint32_t k2a = a1, k2b = b1;

  const f4* x4 = (const f4*)d_in[0];
  float* out = (float*)d_out;
  float* maskf = out + X_ELEMS;                // tuple output #2

  uint32_t* rng = (uint32_t*)d_ws;             // 48000 u32
  uint32_t* mw  = rng + NDRAW;                 // 32*48 packed mask words

  rng_kernel<<<(NPAIR + 255) / 256, 256, 0, stream>>>(rng, k1a, k1b, k2a, k2b);
  mask_kernel<<<MASK_B, 32, 0, stream>>>(rng, mw, maskf);
  apply_async_kernel<<<1024, APPLY_WAVES_PER_BLOCK * 32, 0, stream>>>(x4, (f4*)out, mw);
}